// ShiftedWindowMSA_91164975825301
// MI455X (gfx1250) — compile-verified
//
#include <hip/hip_runtime.h>
#include <math.h>

// ---------------------------------------------------------------------------
// Fused Swin shifted-window MSA for gfx1250 (MI455X), fp32 via
// V_WMMA_F32_16X16X4_F32. One block = 7 windows (half a window row) of one
// batch image. Per head: stage W[:,96h:96h+96) in K=64 chunks through LDS
// in a k-pair-interleaved layout (B fragment = one aligned ds_load_b64,
// bank-conflict-free), WMMA the 112x96 qkv tile (bias-seeded accumulators),
// then do fully wave-local windowed attention (QK^T -> mask -> softmax -> AV)
// with WMMA, scattering the output with the reverse cyclic shift.
// ---------------------------------------------------------------------------

typedef __attribute__((ext_vector_type(2))) float v2f;
typedef __attribute__((ext_vector_type(8))) float v8f;

#define NH        8
#define HD        32
#define DMODEL    256
#define HGRID     56
#define MWIN      7       // windows per block
#define MTOK      112     // tokens per block (7 windows * 16)
#define KCHUNK    64      // K chunk staged in LDS per pass
#define XS_STRIDE 68      // 64 + 4   (stride mod 64 == 4  -> conflict-free A frags)
#define WP_STRIDE 224     // 2*96+32  (stride mod 64 == 32 -> lane halves on disjoint banks)
#define NPAIR     (KCHUNK / 2)   // 32 k-pairs per chunk
#define QK_STRIDE 100     // 96 + 4   (8*stride mod 64 == 32 -> conflict-free C stores)
#define ATT_STRIDE 18

#define XS_FLOATS  (MTOK * XS_STRIDE)        // 7616
#define WH_FLOATS  (NPAIR * WP_STRIDE)       // 7168
#define QK_FLOATS  (MTOK * QK_STRIDE)        // 11200
#define ATT_FLOATS (8 * 16 * ATT_STRIDE)     // 2304
#define SMEM_FLOATS (XS_FLOATS + WH_FLOATS + QK_FLOATS + ATT_FLOATS)
#define SMEM_BYTES  (SMEM_FLOATS * 4 + MTOK * 4)   // + rowoff[112] ints

static __device__ __forceinline__ v8f wmma_f32_4(v2f a, v2f b, v8f c) {
    return __builtin_amdgcn_wmma_f32_16x16x4_f32(false, a, false, b,
                                                 (short)0, c, false, false);
}

__global__ __launch_bounds__(256)
void swin_msa_fused(const float* __restrict__ x, const float* __restrict__ Wg,
                    const float* __restrict__ bvec, float* __restrict__ out) {
    extern __shared__ float smem[];
    float* xs  = smem;                    // [MTOK][XS_STRIDE]
    float* whp = xs + XS_FLOATS;          // [NPAIR][WP_STRIDE] pair-interleaved
    float* qk  = whp + WH_FLOATS;         // [MTOK][QK_STRIDE]
    float* att = qk + QK_FLOATS;          // [8 waves][16][ATT_STRIDE]
    int* rowoff = (int*)(att + ATT_FLOATS); // [MTOK] global token rows

    const int half = blockIdx.x;          // 0/1 -> windows j = half*7 + w
    const int wi   = blockIdx.y;          // window row 0..13
    const int bb   = blockIdx.z;          // batch
    const int tid  = threadIdx.x;
    const int wave = tid >> 5;
    const int lane = tid & 31;
    const int r    = lane & 15;
    const int hiK  = (lane >> 4) << 1;    // 0 for lanes 0..15, 2 for 16..31
    const int hiP  = lane >> 4;           // k-pair select for B fragments
    const int rowb = (lane >= 16) ? 8 : 0;

    // token -> global row map (cyclic shift by +2 applied; same map is used
    // to gather x and to scatter out: roll(-2) in / roll(+2) out hit the
    // same physical rows).
    if (tid < MTOK) {
        int w  = tid >> 4;
        int t  = tid & 15;
        int gh = 4 * wi + (t >> 2) + 2;               if (gh >= HGRID) gh -= HGRID;
        int gw = 4 * (half * MWIN + w) + (t & 3) + 2; if (gw >= HGRID) gw -= HGRID;
        rowoff[tid] = bb * (HGRID * HGRID) + gh * HGRID + gw;
    }
    __syncthreads();

    const bool lastRow = (wi == 13);
    const bool lastCol = (half == 1) && (wave == 6);
    // row mask: -inf where (t>=8) != (s>=8); t>=8 <=> high lane half, s = lane&15
    const bool maskR = lastRow && (((lane >= 16) ? 1 : 0) != ((lane & 8) ? 1 : 0));
    const float scale = 0.17677669529663687f;     // 1/sqrt(32)

    for (int h = 0; h < NH; ++h) {
        v8f acc[6];
        if (wave < MWIN) {
            const float* bp = bvec + 96 * h;
            #pragma unroll
            for (int n = 0; n < 6; ++n) {
                float bv = bp[16 * n + r];
                v8f t;
                #pragma unroll
                for (int q = 0; q < 8; ++q) t[q] = bv;
                acc[n] = t;
            }
        }

        // ---- qkv GEMM over K=256 in chunks of 64, staged through LDS ----
        for (int kc = 0; kc < 4; ++kc) {
            __syncthreads();   // previous pass done reading xs/whp
            // stage x chunk: 112 rows x 64 floats (gathered, shifted rows)
            #pragma unroll
            for (int it = 0; it < 7; ++it) {
                int idx = tid + 256 * it;           // < 1792
                int row = idx >> 4;
                int c4  = (idx & 15) << 2;
                const float4 v = *(const float4*)(x + (size_t)rowoff[row] * DMODEL
                                                    + kc * KCHUNK + c4);
                *(float4*)&xs[row * XS_STRIDE + c4] = v;
            }
            // stage W chunk, k-pair interleaved: whp[p][2n+(k&1)] = W[2p+k&1][n].
            // Each unit: two rows' float2 -> one interleaved float4 LDS store.
            #pragma unroll
            for (int it = 0; it < 6; ++it) {
                int idx  = tid + 256 * it;          // < 1536 = 32 pairs * 48 col2
                int pair = idx / 48;
                int c2g  = idx % 48;
                const float* g0 = Wg + (size_t)(kc * KCHUNK + 2 * pair) * (3 * DMODEL)
                                     + 96 * h + 2 * c2g;
                float2 a0 = *(const float2*)g0;
                float2 a1 = *(const float2*)(g0 + 3 * DMODEL);
                float4 st; st.x = a0.x; st.y = a1.x; st.z = a0.y; st.w = a1.y;
                *(float4*)&whp[pair * WP_STRIDE + 4 * c2g] = st;
            }
            __syncthreads();   // chunk ready

            if (wave < MWIN) {
                const float* xrow = &xs[(16 * wave + r) * XS_STRIDE + hiK];
                const float* wrow = &whp[hiP * WP_STRIDE + 2 * r];
                #pragma unroll 4
                for (int k0 = 0; k0 < KCHUNK; k0 += 4) {
                    // group all fragment loads, then all WMMAs (one dscnt wait)
                    v2f a = *(const v2f*)(xrow + k0);      // A[m=r][k0+hiK, +1]
                    v2f bf[6];
                    #pragma unroll
                    for (int n = 0; n < 6; ++n)            // B[k pair][n=16n+r]
                        bf[n] = *(const v2f*)(wrow + (k0 >> 1) * WP_STRIDE + 32 * n);
                    #pragma unroll
                    for (int n = 0; n < 6; ++n)
                        acc[n] = wmma_f32_4(a, bf[n], acc[n]);
                }
            }
        }

        if (wave < MWIN) {
            // ---- spill qkv tile to LDS (wave-local rows; no barrier needed) ----
            int qrowb = 16 * wave + rowb;
            #pragma unroll
            for (int n = 0; n < 6; ++n)
                #pragma unroll
                for (int v = 0; v < 8; ++v)
                    qk[(qrowb + v) * QK_STRIDE + 16 * n + r] = acc[n][v];

            // ---- scores = Q K^T  (qkv cols interleaved: Q=3e, K=3e+1, V=3e+2)
            const float* qrow = &qk[(16 * wave + r) * QK_STRIDE];
            v8f s = {0.f, 0.f, 0.f, 0.f, 0.f, 0.f, 0.f, 0.f};
            #pragma unroll
            for (int k0 = 0; k0 < HD; k0 += 4) {
                int ka = 3 * (k0 + hiK);
                v2f a, bf;
                a.x  = qrow[ka];     a.y  = qrow[ka + 3];      // Q[r][k], Q[r][k+1]
                bf.x = qrow[ka + 1]; bf.y = qrow[ka + 4];      // K[r][k], K[r][k+1]
                s = wmma_f32_4(a, bf, s);
            }

            // ---- scale + shift masks + row softmax (rows live across 16 lanes)
            v8f p;
            #pragma unroll
            for (int v = 0; v < 8; ++v) {
                float xv = s[v] * scale;
                bool mC = lastCol && (((v & 2) != 0) != ((lane & 2) != 0));
                if (maskR || mC) xv = -1e30f;
                float m = xv;
                #pragma unroll
                for (int off = 8; off; off >>= 1)
                    m = fmaxf(m, __shfl_xor(m, off, 16));
                float e = __expf(xv - m);
                float sm = e;
                #pragma unroll
                for (int off = 8; off; off >>= 1)
                    sm += __shfl_xor(sm, off, 16);
                p[v] = e / sm;
            }

            // ---- stage att through LDS to convert C-layout -> A-layout ----
            float* am = &att[wave * 16 * ATT_STRIDE];
            #pragma unroll
            for (int v = 0; v < 8; ++v)
                am[(rowb + v) * ATT_STRIDE + r] = p[v];

            // ---- out = att @ V  (16x16 @ 16x32) ----
            v8f o0 = {0.f, 0.f, 0.f, 0.f, 0.f, 0.f, 0.f, 0.f};
            v8f o1 = {0.f, 0.f, 0.f, 0.f, 0.f, 0.f, 0.f, 0.f};
            #pragma unroll
            for (int k0 = 0; k0 < 16; k0 += 4) {
                v2f a = *(const v2f*)(am + r * ATT_STRIDE + k0 + hiK);
                const float* v0p = &qk[(16 * wave + k0 + hiK) * QK_STRIDE + 2];
                const float* v1p = v0p + QK_STRIDE;
                v2f b0; b0.x = v0p[3 * r];        b0.y = v1p[3 * r];
                o0 = wmma_f32_4(a, b0, o0);
                v2f b1; b1.x = v0p[3 * (16 + r)]; b1.y = v1p[3 * (16 + r)];
                o1 = wmma_f32_4(a, b1, o1);
            }

            // ---- scatter with reverse roll (same rowoff map), NT stores ----
            float* ob = out + 32 * h + r;
            #pragma unroll
            for (int v = 0; v < 8; ++v) {
                int grow = rowoff[16 * wave + rowb + v];
                float* pdst = ob + (size_t)grow * DMODEL;
                __builtin_nontemporal_store(o0[v], pdst);
                __builtin_nontemporal_store(o1[v], pdst + 16);
            }
        }
    }
}

extern "C" void kernel_launch(void* const* d_in, const int* in_sizes, int n_in,
                              void* d_out, int out_size, void* d_ws, size_t ws_size,
                              hipStream_t stream) {
    (void)in_sizes; (void)n_in; (void)d_ws; (void)ws_size; (void)out_size;
    const float* x = (const float*)d_in[0];
    const float* W = (const float*)d_in[1];
    const float* b = (const float*)d_in[2];
    float* out = (float*)d_out;

    static_assert(SMEM_BYTES < 320 * 1024, "LDS budget");
    hipFuncSetAttribute((const void*)swin_msa_fused,
                        hipFuncAttributeMaxDynamicSharedMemorySize, SMEM_BYTES);

    dim3 grid(2, 14, 32);   // half-row, window-row, batch
    swin_msa_fused<<<grid, 256, SMEM_BYTES, stream>>>(x, W, b, out);
}